// GCN_69569880261297
// MI455X (gfx1250) — compile-verified
//
#include <hip/hip_runtime.h>
#include <hip/hip_bf16.h>

typedef __attribute__((ext_vector_type(2))) float v2f;
typedef __attribute__((ext_vector_type(8))) float v8f;

#define TPB 256

// ---------------- preprocessing kernels ----------------

__global__ void zero_f32(float* __restrict__ p, int n) {
    int t = blockIdx.x * blockDim.x + threadIdx.x;
    if (t < n) p[t] = 0.0f;
}

// Build int32 src/dst (edges + self loops) and accumulate degree of dst.
__global__ void build_edges(const long long* __restrict__ ei,
                            int* __restrict__ src, int* __restrict__ dst,
                            float* __restrict__ deg, int n_edges, int n_total) {
    int e = blockIdx.x * blockDim.x + threadIdx.x;
    if (e >= n_total) return;
    int s, d;
    if (e < n_edges) { s = (int)ei[e]; d = (int)ei[n_edges + e]; }
    else             { s = d = e - n_edges; }
    src[e] = s;
    dst[e] = d;
    atomicAdd(&deg[d], 1.0f);
}

__global__ void compute_dinv(float* __restrict__ dinv, int n) {
    int i = blockIdx.x * blockDim.x + threadIdx.x;
    if (i >= n) return;
    float d = dinv[i];
    dinv[i] = (d > 0.0f) ? rsqrtf(d) : 0.0f;
}

__global__ void compute_norm(const int* __restrict__ src, const int* __restrict__ dst,
                             const float* __restrict__ dinv, float* __restrict__ norm,
                             int n_total) {
    int e = blockIdx.x * blockDim.x + threadIdx.x;
    if (e >= n_total) return;
    norm[e] = dinv[src[e]] * dinv[dst[e]];
}

// h0: N x 32 (stride 32), first 2 cols = x, rest zero.
__global__ void init_h(const float* __restrict__ x, float* __restrict__ h, int n32) {
    int t = blockIdx.x * blockDim.x + threadIdx.x;
    if (t >= n32) return;
    int i = t >> 5, c = t & 31;
    h[t] = (c < 2) ? x[i * 2 + c] : 0.0f;
}

// Zero-pad W (fin x fout) into 32x32, and b into 32.
__global__ void pad_wb(const float* __restrict__ W, const float* __restrict__ b,
                       float* __restrict__ Wp, float* __restrict__ bp,
                       int fin, int fout) {
    int t = blockIdx.x * blockDim.x + threadIdx.x;
    if (t >= 1024) return;
    int i = t >> 5, j = t & 31;
    Wp[t] = (i < fin && j < fout) ? W[i * fout + j] : 0.0f;
    if (t < 32) bp[t] = (t < fout) ? b[t] : 0.0f;
}

// ---------------- GEMM: HW = H(Nx32) @ Wp(32x32), f32 WMMA ----------------
// One wave computes a 16-row x 16-col tile, K-loop of 8 x V_WMMA_F32_16X16X4_F32.

__global__ void wmma_matmul32(const float* __restrict__ H, const float* __restrict__ Wp,
                              float* __restrict__ HW, int n_nodes) {
    int wave = (blockIdx.x * blockDim.x + threadIdx.x) >> 5;
    int lane = threadIdx.x & 31;
    int n_tiles = (n_nodes >> 4) * 2;     // row tiles * 2 column tiles
    if (wave >= n_tiles) return;          // wave-uniform exit: EXEC all-ones at WMMA

    int row_base = (wave >> 1) * 16;
    int col_base = (wave & 1) * 16;
    int half = lane >> 4;                 // 0: lanes 0-15, 1: lanes 16-31
    int l15  = lane & 15;

    const float* hrow = H + (size_t)(row_base + l15) * 32;
    v8f acc = {};
#pragma unroll
    for (int k = 0; k < 32; k += 4) {
        // A 16x4 (M x K): lane<16 -> K k,k+1 ; lane>=16 -> K k+2,k+3
        v2f a, b;
        a.x = hrow[k + 2 * half + 0];
        a.y = hrow[k + 2 * half + 1];
        // B 4x16 (K x N): N = col_base + l15 striped across lanes
        b.x = Wp[(k + 2 * half + 0) * 32 + col_base + l15];
        b.y = Wp[(k + 2 * half + 1) * 32 + col_base + l15];
        acc = __builtin_amdgcn_wmma_f32_16x16x4_f32(
            /*neg_a=*/false, a, /*neg_b=*/false, b,
            /*c_mod=*/(short)0, acc, /*reuse_a=*/false, /*reuse_b=*/false);
    }
    // D 16x16: VGPR r -> row r (lanes 0-15) / row r+8 (lanes 16-31)
#pragma unroll
    for (int r = 0; r < 8; ++r)
        HW[(size_t)(row_base + r + 8 * half) * 32 + col_base + l15] = acc[r];
}

// ---------------- edge scatter: acc[dst] += HW[src] * norm ----------------
// grid.y indexes a 4-column slice; 128-bit gather + 4 f32 atomics (L2-resident).

__global__ void scatter_edges(const float* __restrict__ HW, const int* __restrict__ src,
                              const int* __restrict__ dst, const float* __restrict__ norm,
                              float* __restrict__ ACC, int n_total) {
    int e = blockIdx.x * blockDim.x + threadIdx.x;
    if (e >= n_total) return;
    int c4 = blockIdx.y << 2;
    float nv = norm[e];
    const float4 m = *(const float4*)(HW + (size_t)src[e] * 32 + c4);
    float* o = ACC + (size_t)dst[e] * 32 + c4;
    atomicAdd(o + 0, m.x * nv);
    atomicAdd(o + 1, m.y * nv);
    atomicAdd(o + 2, m.z * nv);
    atomicAdd(o + 3, m.w * nv);
}

// ---------------- epilogues ----------------

__global__ void bias_relu(const float* __restrict__ acc, const float* __restrict__ bp,
                          float* __restrict__ h, int n32, int fout) {
    int t = blockIdx.x * blockDim.x + threadIdx.x;
    if (t >= n32) return;
    int c = t & 31;
    float v = (c < fout) ? acc[t] + bp[c] : 0.0f;
    h[t] = fmaxf(v, 0.0f);
}

__global__ void finalize(const float* __restrict__ acc, const float* __restrict__ bp,
                         const float* __restrict__ x, float* __restrict__ out, int n2) {
    int t = blockIdx.x * blockDim.x + threadIdx.x;
    if (t >= n2) return;
    int i = t >> 1, c = t & 1;
    out[t] = acc[(size_t)i * 32 + c] + bp[c] + x[t];
}

// ---------------- host orchestration ----------------

extern "C" void kernel_launch(void* const* d_in, const int* in_sizes, int n_in,
                              void* d_out, int out_size, void* d_ws, size_t ws_size,
                              hipStream_t stream) {
    const float*     x  = (const float*)d_in[0];
    const long long* ei = (const long long*)d_in[1];
    const int N  = in_sizes[0] / 2;   // 100000 nodes (x is N x 2)
    const int E  = in_sizes[1] / 2;   // 1.6M edges
    const int ET = E + N;             // + self loops
    const int N32 = N * 32;

    char* ws = (char*)d_ws;
    size_t off = 0;
    auto alloc = [&](size_t bytes) -> void* {
        void* p = ws + off;
        off += (bytes + 255) & ~(size_t)255;
        return p;
    };
    int*   src32 = (int*)  alloc((size_t)ET * 4);
    int*   dst32 = (int*)  alloc((size_t)ET * 4);
    float* norm  = (float*)alloc((size_t)ET * 4);
    float* dinv  = (float*)alloc((size_t)N * 4);      // holds deg, then rsqrt(deg)
    float* hA    = (float*)alloc((size_t)N32 * 4);
    float* hw    = (float*)alloc((size_t)N32 * 4);
    float* acc   = (float*)alloc((size_t)N32 * 4);
    float* Wpad  = (float*)alloc(8 * 1024 * 4);
    float* bpad  = (float*)alloc(8 * 32 * 4);

    static const int fin[8]  = {2, 16, 32, 32, 32, 32, 32, 16};
    static const int fout[8] = {16, 32, 32, 32, 32, 32, 16, 2};

    // Preprocess: degrees, dinv, int32 edges, per-edge norm, h0, padded weights
    zero_f32<<<(N + TPB - 1) / TPB, TPB, 0, stream>>>(dinv, N);
    build_edges<<<(ET + TPB - 1) / TPB, TPB, 0, stream>>>(ei, src32, dst32, dinv, E, ET);
    compute_dinv<<<(N + TPB - 1) / TPB, TPB, 0, stream>>>(dinv, N);
    compute_norm<<<(ET + TPB - 1) / TPB, TPB, 0, stream>>>(src32, dst32, dinv, norm, ET);
    init_h<<<(N32 + TPB - 1) / TPB, TPB, 0, stream>>>(x, hA, N32);
    for (int l = 0; l < 8; ++l) {
        const float* W = (const float*)d_in[2 + 2 * l];
        const float* b = (const float*)d_in[3 + 2 * l];
        pad_wb<<<4, TPB, 0, stream>>>(W, b, Wpad + l * 1024, bpad + l * 32, fin[l], fout[l]);
    }

    // 8 GCN layers
    const int n_waves   = (N >> 4) * 2;
    const int mm_blocks = (n_waves * 32 + TPB - 1) / TPB;
    for (int l = 0; l < 8; ++l) {
        wmma_matmul32<<<mm_blocks, TPB, 0, stream>>>(hA, Wpad + l * 1024, hw, N);
        zero_f32<<<(N32 + TPB - 1) / TPB, TPB, 0, stream>>>(acc, N32);
        dim3 g((ET + TPB - 1) / TPB, (fout[l] + 3) / 4);
        scatter_edges<<<g, TPB, 0, stream>>>(hw, src32, dst32, norm, acc, ET);
        if (l < 7) {
            bias_relu<<<(N32 + TPB - 1) / TPB, TPB, 0, stream>>>(acc, bpad + l * 32, hA, N32, fout[l]);
        } else {
            finalize<<<(N * 2 + TPB - 1) / TPB, TPB, 0, stream>>>(acc, bpad + l * 32, x, (float*)d_out, N * 2);
        }
    }
}